// GCNWithEdgeWeights_12472585027793
// MI455X (gfx1250) — compile-verified
//
#include <hip/hip_runtime.h>
#include <hip/hip_bf16.h>

#define N_NODES_C  50000
#define N_EDGES_C  400000
#define EMB_C      16
#define MLP_H_C    8
#define EDGE_F_C   8
#define N_GRAPHS_C 64

typedef float v2f __attribute__((ext_vector_type(2)));
typedef float v8f __attribute__((ext_vector_type(8)));

// Order-preserving float<->uint encoding so segment-max can use integer atomics.
__device__ __forceinline__ unsigned f2o(float f) {
    unsigned u = __float_as_uint(f);
    return (u & 0x80000000u) ? ~u : (u | 0x80000000u);
}
__device__ __forceinline__ float o2f(unsigned v) {
    unsigned u = (v & 0x80000000u) ? (v ^ 0x80000000u) : ~v;
    return __uint_as_float(u);
}
__device__ __forceinline__ bool finitef(float v) {
    return ((__float_as_uint(v) >> 23) & 0xFFu) != 0xFFu;
}
__device__ __forceinline__ float relu6f(float v) { return fminf(fmaxf(v, 0.f), 6.f); }

// ---------------------------------------------------------------------------
// Edge kernel: one wave handles 16 edges.
//   h = relu(edge_attr @ mw1 + mb1)                      (lanes 0-15, VALU)
//   W = h @ mw2 + mb2  via V_WMMA_F32_16X16X4_F32        (M=16 edges, 16 N-tiles, 2 K-steps)
//   msg[e][o] = sum_i x[src[e]][i] * W[e][i][o]          (tile i == N-tile nt; VALU FMA)
//   scatter: atomic max (encoded) into agg[dst][o]
// ---------------------------------------------------------------------------
__global__ __launch_bounds__(256) void nnconv_edge_kernel(
    const float* __restrict__ xin, const int* __restrict__ srcI, const int* __restrict__ dstI,
    const float* __restrict__ ea, const float* __restrict__ mw1, const float* __restrict__ mb1,
    const float* __restrict__ mw2, const float* __restrict__ mb2,
    unsigned* __restrict__ agg, int nE)
{
    __shared__ float sx[8][16][17];  // x[src] per edge-in-group (padded: 17*l%64 distinct)
    __shared__ float sh[8][16][9];   // edge-MLP output h
    __shared__ int   sd[8][16];      // dst node ids

    const int lane = threadIdx.x & 31;
    const int wave = threadIdx.x >> 5;
    const int hi   = lane >> 4;      // lane half
    const int l16  = lane & 15;
    const int eBase = (blockIdx.x * 8 + wave) * 16;

    if (hi == 0) {
        int e = eBase + l16; if (e >= nE) e = nE - 1;
        sd[wave][l16] = dstI[e];
        const float4* xr = (const float4*)(xin + (size_t)srcI[e] * EMB_C);
        float4 q0 = xr[0], q1 = xr[1], q2 = xr[2], q3 = xr[3];
        sx[wave][l16][0]=q0.x;  sx[wave][l16][1]=q0.y;  sx[wave][l16][2]=q0.z;  sx[wave][l16][3]=q0.w;
        sx[wave][l16][4]=q1.x;  sx[wave][l16][5]=q1.y;  sx[wave][l16][6]=q1.z;  sx[wave][l16][7]=q1.w;
        sx[wave][l16][8]=q2.x;  sx[wave][l16][9]=q2.y;  sx[wave][l16][10]=q2.z; sx[wave][l16][11]=q2.w;
        sx[wave][l16][12]=q3.x; sx[wave][l16][13]=q3.y; sx[wave][l16][14]=q3.z; sx[wave][l16][15]=q3.w;
        const float4* er = (const float4*)(ea + (size_t)e * EDGE_F_C);
        float4 e0 = er[0], e1 = er[1];
        float a[EDGE_F_C] = {e0.x, e0.y, e0.z, e0.w, e1.x, e1.y, e1.z, e1.w};
        #pragma unroll
        for (int j = 0; j < MLP_H_C; ++j) {
            float hv = mb1[j];
            #pragma unroll
            for (int k = 0; k < EDGE_F_C; ++k) hv = fmaf(a[k], mw1[k * MLP_H_C + j], hv);
            sh[wave][l16][j] = fmaxf(hv, 0.f);
        }
    }

    // B fragments of mw2 (8x256) + bias mb2 (256).
    // B (4x16 KxN) layout mirrors documented A layout: VGPR0 = K {0|2} per half, VGPR1 = K {1|3}.
    const int k0 = hi ? 2 : 0;
    float b0[16][2], b1[16][2], biasn[16];
    #pragma unroll
    for (int nt = 0; nt < 16; ++nt) {
        int n = nt * 16 + l16;
        biasn[nt] = mb2[n];
        #pragma unroll
        for (int s = 0; s < 2; ++s) {
            b0[nt][s] = mw2[(s * 4 + k0) * 256 + n];
            b1[nt][s] = mw2[(s * 4 + k0 + 1) * 256 + n];
        }
    }

    __syncthreads();

    // A fragments (16x4 f32, two K-steps covering K=0..7)
    v2f A0, A1;
    A0.x = sh[wave][l16][k0];     A0.y = sh[wave][l16][k0 + 1];
    A1.x = sh[wave][l16][4 + k0]; A1.y = sh[wave][l16][4 + k0 + 1];

    float msgacc[8];
    #pragma unroll
    for (int r = 0; r < 8; ++r) msgacc[r] = 0.f;

    #pragma unroll
    for (int nt = 0; nt < 16; ++nt) {
        v8f C;
        #pragma unroll
        for (int r = 0; r < 8; ++r) C[r] = biasn[nt];     // seed with mb2 bias
        v2f B0; B0.x = b0[nt][0]; B0.y = b1[nt][0];
        v2f B1; B1.x = b0[nt][1]; B1.y = b1[nt][1];
        v8f D = __builtin_amdgcn_wmma_f32_16x16x4_f32(false, A0, false, B0, (short)0, C, false, false);
        D     = __builtin_amdgcn_wmma_f32_16x16x4_f32(false, A1, false, B1, (short)0, D, false, false);
        // D tile nt == W[e, i=nt, 0..15]; lane holds (o=l16, e=8*hi+r in VGPR r)
        #pragma unroll
        for (int r = 0; r < 8; ++r) {
            float xv = sx[wave][8 * hi + r][nt];          // broadcast within lane half
            msgacc[r] = fmaf(xv, D[r], msgacc[r]);
        }
    }

    #pragma unroll
    for (int r = 0; r < 8; ++r) {
        int eg = eBase + 8 * hi + r;
        if (eg < nE) {
            int d = sd[wave][8 * hi + r];
            atomicMax(agg + (size_t)d * EMB_C + l16, f2o(msgacc[r]));
        }
    }
}

// h_out = relu6( decode(agg) + x @ root + bias )
__global__ __launch_bounds__(256) void nnconv_node_kernel(
    const unsigned* __restrict__ agg, const float* __restrict__ xin,
    const float* __restrict__ root, const float* __restrict__ bias,
    float* __restrict__ hout, int nN)
{
    int n = blockIdx.x * blockDim.x + threadIdx.x;
    if (n >= nN) return;
    float xv[EMB_C];
    const float4* xr = (const float4*)(xin + (size_t)n * EMB_C);
    #pragma unroll
    for (int q = 0; q < 4; ++q) {
        float4 v = xr[q];
        xv[q*4+0]=v.x; xv[q*4+1]=v.y; xv[q*4+2]=v.z; xv[q*4+3]=v.w;
    }
    #pragma unroll
    for (int o = 0; o < EMB_C; ++o) {
        float acc = bias[o];
        #pragma unroll
        for (int i = 0; i < EMB_C; ++i) acc = fmaf(xv[i], root[i * EMB_C + o], acc);
        unsigned enc = agg[(size_t)n * EMB_C + o];
        float av = 0.f;
        if (enc != 0u) { float t = o2f(enc); if (finitef(t)) av = t; }
        hout[(size_t)n * EMB_C + o] = relu6f(acc + av);
    }
}

__global__ __launch_bounds__(256) void pool_kernel(
    const float* __restrict__ h, const int* __restrict__ batch,
    unsigned* __restrict__ maxp, float* __restrict__ sump, float* __restrict__ cnt, int nN)
{
    int n = blockIdx.x * blockDim.x + threadIdx.x;
    if (n >= nN) return;
    int b = batch[n];
    #pragma unroll
    for (int o = 0; o < EMB_C; ++o) {
        float v = h[(size_t)n * EMB_C + o];
        atomicMax(maxp + b * EMB_C + o, f2o(v));
        atomicAdd(sump + b * EMB_C + o, v);
    }
    atomicAdd(cnt + b, 1.f);
}

// combined = [maxpool | meanpool | graph_features]; logits = relu(combined@fc1)@fc2
__global__ __launch_bounds__(128) void head_kernel(
    const unsigned* __restrict__ maxp, const float* __restrict__ sump, const float* __restrict__ cnt,
    const float* __restrict__ gfeat, const float* __restrict__ w1, const float* __restrict__ b1,
    const float* __restrict__ w2, const float* __restrict__ b2, float* __restrict__ out)
{
    int g = blockIdx.x, t = threadIdx.x;
    __shared__ float comb[48];
    __shared__ float red[128];
    if (t < 16) {
        unsigned enc = maxp[g * 16 + t];
        float m = 0.f;
        if (enc != 0u) { float q = o2f(enc); if (finitef(q)) m = q; }
        comb[t] = m;
    } else if (t < 32) {
        comb[t] = sump[g * 16 + (t - 16)] / fmaxf(cnt[g], 1.f);
    } else if (t < 48) {
        comb[t] = gfeat[g * 16 + (t - 32)];
    }
    __syncthreads();
    float acc = b1[t];
    #pragma unroll
    for (int k = 0; k < 48; ++k) acc = fmaf(comb[k], w1[k * 128 + t], acc);
    red[t] = fmaxf(acc, 0.f) * w2[t];
    __syncthreads();
    for (int s = 64; s > 0; s >>= 1) {
        if (t < s) red[t] += red[t + s];
        __syncthreads();
    }
    if (t == 0) out[g] = red[0] + b2[0];
    if (t < 48) out[64 + g * 48 + t] = comb[t];
}

extern "C" void kernel_launch(void* const* d_in, const int* in_sizes, int n_in,
                              void* d_out, int out_size, void* d_ws, size_t ws_size,
                              hipStream_t stream)
{
    const float* x    = (const float*)d_in[0];
    const int*   ei   = (const int*)d_in[1];    // int32 (JAX x64 off): [2, E]
    const float* ea   = (const float*)d_in[2];
    const int*   bat  = (const int*)d_in[3];
    const float* gf   = (const float*)d_in[4];
    const float* cw[3][6];
    for (int l = 0; l < 3; ++l)
        for (int q = 0; q < 6; ++q)
            cw[l][q] = (const float*)d_in[5 + l * 6 + q];  // mw1, mb1, mw2, mb2, root, bias
    const float* fc1w = (const float*)d_in[23];
    const float* fc1b = (const float*)d_in[24];
    const float* fc2w = (const float*)d_in[25];
    const float* fc2b = (const float*)d_in[26];

    char* ws = (char*)d_ws;
    unsigned* agg  = (unsigned*)(ws);                       // 50000*16*4 = 3.2 MB
    float*    hA   = (float*)(ws + 3200000);
    float*    hB   = (float*)(ws + 6400000);
    unsigned* maxp = (unsigned*)(ws + 9600000);             // 4 KB
    float*    sump = (float*)(ws + 9600000 + 4096);         // 4 KB
    float*    cnt  = (float*)(ws + 9600000 + 8192);         // 256 B

    const int* srcI = ei;
    const int* dstI = ei + N_EDGES_C;

    dim3 eblk(256), egrd((N_EDGES_C + 127) / 128);          // 8 waves/block, 16 edges/wave
    dim3 nblk(256), ngrd((N_NODES_C + 255) / 256);

    const float* lin = x;
    float* louts[3] = { hA, hB, hA };
    for (int l = 0; l < 3; ++l) {
        hipMemsetAsync(agg, 0, (size_t)N_NODES_C * EMB_C * 4, stream);
        nnconv_edge_kernel<<<egrd, eblk, 0, stream>>>(lin, srcI, dstI, ea,
            cw[l][0], cw[l][1], cw[l][2], cw[l][3], agg, N_EDGES_C);
        nnconv_node_kernel<<<ngrd, nblk, 0, stream>>>(agg, lin, cw[l][4], cw[l][5],
            louts[l], N_NODES_C);
        lin = louts[l];
    }
    hipMemsetAsync(maxp, 0, 4096 + 4096 + 256, stream);
    pool_kernel<<<ngrd, nblk, 0, stream>>>(lin, bat, maxp, sump, cnt, N_NODES_C);
    head_kernel<<<N_GRAPHS_C, 128, 0, stream>>>(maxp, sump, cnt, gf, fc1w, fc1b, fc2w, fc2b,
                                                (float*)d_out);
}